// MultiHeadAttention_80324478370448
// MI455X (gfx1250) — compile-verified
//
#include <hip/hip_runtime.h>

typedef float v2f __attribute__((ext_vector_type(2)));
typedef float v8f __attribute__((ext_vector_type(8)));
typedef unsigned int v4u __attribute__((ext_vector_type(4)));
typedef unsigned int v8u __attribute__((ext_vector_type(8)));

#define BB  2
#define SS  513
#define HH  512
#define NHH 8
#define DD  64
#define LL  9
#define BSR (BB*SS)   // 1026 flattened (b,t) rows

// ---------------------------------------------------------------------------
// TDM helper: async-load a 2D tile (tileRows x 512 f32, constant row stride)
// from global memory into LDS. OOB rows (y >= tensorRows) are zero-filled by
// the TDM, which implements our tile zero-padding for free.
// D# group0: [1:0]=count=1, [63:32]=lds_addr, [120:64]=global_addr,
//            [127:126]=type=2.  D# group1: dims/strides (2D -> groups 2/3 NULL).
// ---------------------------------------------------------------------------
__device__ __forceinline__ void tdm_load_rows_f32(
    const float* gsrc, void* ldst, int tileRows, int tensorRows,
    unsigned int rowStrideElems)
{
  if (tensorRows < 0) tensorRows = 0;
  unsigned long long ga = (unsigned long long)(uintptr_t)gsrc;
  unsigned int lo = (unsigned int)(uintptr_t)ldst;  // low 32 bits = LDS offset
  v4u g0;
  g0[0] = 1u;                                        // count=1, user mode
  g0[1] = lo;                                        // lds_addr
  g0[2] = (unsigned int)(ga & 0xffffffffu);          // global_addr[31:0]
  g0[3] = (unsigned int)((ga >> 32) & 0x01ffffffu)   // global_addr[56:32]
        | (2u << 30);                                // type=2 ("image")
  const unsigned int td0 = HH;                       // tensor_dim0 = 512 elems
  const unsigned int td1 = (unsigned int)tensorRows; // valid rows (OOB -> 0)
  v8u g1;
  g1[0] = (2u << 16);                                // data_size=2 (4B), mask=0
  g1[1] = (td0 & 0xffffu) << 16;                     // tensor_dim0[15:0]
  g1[2] = ((td0 >> 16) & 0xffffu) | ((td1 & 0xffffu) << 16);
  g1[3] = ((td1 >> 16) & 0xffffu) | ((unsigned int)HH << 16); // tile_dim0=512
  g1[4] = (unsigned int)tileRows;                    // tile_dim1, tile_dim2=0
  g1[5] = rowStrideElems;                            // tensor_dim0_stride[31:0]
  g1[6] = 0u;
  g1[7] = 0u;
  asm volatile("tensor_load_to_lds %0, %1" :: "s"(g0), "s"(g1) : "memory");
}

// ---------------------------------------------------------------------------
// Projection GEMM via V_WMMA_F32_16X16X4_F32 (fp32 WMMA, matches f32 ref).
//
// nf==5 ("classed" mode, grid.x = 72): the position-class masks partition
//   tokens into 18 arithmetic subsequences (per b: classes 0,1,2 at t=9j+c;
//   classes 3,4 each split into 3 subsequences t=9j+(3+2m)/(4+2m)), all with
//   constant row stride 9. Gathering rows by subsequence makes every 16-row
//   M-tile single-class -> ONE un-masked GEMM pass (4.5x fewer WMMA than the
//   masked 5-pass formulation). The strided row gather is done by the TDM.
// nf==1 (plain GEMM, grid.x = 65): output projection with Wo.
//
// One wave owns a 16x16 f32 tile; K=512 in steps of 4.
// C/D layout: VGPR j -> (M = j + 8*(lane>=16), N = lane&15).
// ---------------------------------------------------------------------------
__global__ __launch_bounds__(128) void proj_wmma_kernel(
    const float* __restrict__ X, const float* __restrict__ W,
    const float* __restrict__ Bias, float* __restrict__ Y, int nf)
{
  __shared__ float sx[16][HH];   // 32 KB A-tile (16 gathered rows)
  const int tid = threadIdx.x;
  const int wave = tid >> 5, lane = tid & 31;
  const int hi = lane >> 4, lo = lane & 15;

  int baserow, rs, nvalid;
  const float* wmat;
  if (nf == 5) {
    int sub  = blockIdx.x >> 2;          // subsequence 0..17
    int j0   = (blockIdx.x & 3) * 16;    // row offset within subsequence
    int b    = sub / 9; sub -= b * 9;    // per-b subsequence 0..8
    int cls  = (sub < 3) ? sub : (sub < 6 ? 3 : 4);
    int m    = (sub < 3) ? 0   : (sub - 3) % 3;
    int pos  = (cls < 3) ? cls : ((cls == 3) ? 3 + 2 * m : 4 + 2 * m);
    baserow  = b * SS + pos + 9 * j0;    // global row of tile row 0
    rs       = 9;                        // row stride (rows)
    nvalid   = 57 - j0;                  // valid tile rows (j < 57)
    wmat     = W + (size_t)cls * HH * HH;
  } else {
    baserow  = blockIdx.x * 16;
    rs       = 1;
    nvalid   = BSR - baserow;
    wmat     = W;
  }

  // Each wave TDM-loads its 4 rows of the A tile (async DMA -> LDS).
  {
    const float* gsrc = X + (size_t)(baserow + rs * 4 * wave) * HH;
    tdm_load_rows_f32(gsrc, &sx[4 * wave][0], 4, nvalid - 4 * wave,
                      (unsigned int)rs * HH);
    __builtin_amdgcn_s_wait_tensorcnt(0);
  }
  __syncthreads();

  const int o0 = (blockIdx.y * 4 + wave) * 16;   // N tile

  float bsum = 0.f;
  for (int f = 0; f < nf; ++f) bsum += Bias[f * HH + o0 + lo];
  v8f c;
#pragma unroll
  for (int j = 0; j < 8; ++j) c[j] = bsum;       // bias depends only on column

  const int kof = 2 * hi;
  const float* wrow = wmat + (size_t)(o0 + lo) * HH + kof;
  const float* arow = &sx[lo][kof];
  for (int k0 = 0; k0 < HH; k0 += 4) {
    v2f a, b;
    a[0] = arow[k0];
    a[1] = arow[k0 + 1];
    float2 wv = *(const float2*)(wrow + k0);
    b[0] = wv.x; b[1] = wv.y;
    c = __builtin_amdgcn_wmma_f32_16x16x4_f32(false, a, false, b,
                                              (short)0, c, false, false);
  }
#pragma unroll
  for (int j = 0; j < 8; ++j) {
    int r = j + 8 * hi;
    if (r < nvalid) Y[(size_t)(baserow + rs * r) * HH + o0 + lo] = c[j];
  }
}

// ---------------------------------------------------------------------------
// Per-(b,h,d) sum of v over all keys (for degenerate fully-masked rows).
// ---------------------------------------------------------------------------
__global__ __launch_bounds__(256) void vsum_kernel(
    const float* __restrict__ V, float* __restrict__ VS)
{
  int idx = blockIdx.x * 256 + threadIdx.x;
  if (idx >= BB * HH) return;
  int b = idx / HH, hd = idx % HH;
  float s = 0.f;
  for (int k = 0; k < SS; ++k) s += V[((size_t)b * SS + k) * HH + hd];
  VS[idx] = s;
}

// ---------------------------------------------------------------------------
// Dense attention rows (q < 3: keep-mask all true). One block per (b,q),
// 8 waves = 8 heads. scores[h,k] = (q_h + eq_k) . (k_{h,k} + ek_k).
// ---------------------------------------------------------------------------
__global__ __launch_bounds__(256) void attn_dense_kernel(
    const float* __restrict__ Q, const float* __restrict__ K,
    const float* __restrict__ V, const unsigned char* __restrict__ G,
    const float* __restrict__ EK, const float* __restrict__ EV,
    const float* __restrict__ EQ, float* __restrict__ Xout)
{
  __shared__ float sq[HH];
  __shared__ float sek[32][DD];
  __shared__ float seq[32][DD];
  __shared__ float ssc[NHH][SS + 7];
  const int tid = threadIdx.x;
  const int b = blockIdx.x / 3, qi = blockIdx.x % 3;
  const size_t bq = (size_t)b * SS + qi;

  for (int v = tid; v < HH / 4; v += 256)
    *(float4*)(sq + v * 4) = *(const float4*)(Q + bq * HH + v * 4);

  for (int k0 = 0; k0 < SS; k0 += 32) {
    __syncthreads();
    for (int v = tid; v < 512; v += 256) {       // 32x64 floats of ek and eq
      int kk = v >> 4, d4 = (v & 15) << 2, k = k0 + kk;
      float4 a = make_float4(0.f, 0.f, 0.f, 0.f);
      float4 e = make_float4(0.f, 0.f, 0.f, 0.f);
      if (k < SS) {
        a = *(const float4*)(EK + (bq * SS + k) * DD + d4);
        e = *(const float4*)(EQ + (((size_t)b * SS + k) * SS + qi) * DD + d4);
      }
      *(float4*)(&sek[kk][d4]) = a;
      *(float4*)(&seq[kk][d4]) = e;
    }
    __syncthreads();
    const int h = tid >> 5, kk = tid & 31, k = k0 + kk;
    if (k < SS) {
      const float* krow = K + ((size_t)(b * SS + k) * NHH + h) * DD;
      const float* qh = sq + h * DD;
      float s = 0.f;
#pragma unroll 8
      for (int d = 0; d < DD; ++d)
        s += (qh[d] + seq[kk][d]) * (krow[d] + sek[kk][d]);
      bool m = (G[((size_t)b * SS + qi) * SS + k] != 0);
      ssc[h][k] = m ? s * 0.125f : -100000.f;    // scale = 1/sqrt(64)
    }
  }
  __syncthreads();

  const int h = tid >> 5, lane = tid & 31;
  float mx = -3.4e38f;
  for (int k = lane; k < SS; k += 32) mx = fmaxf(mx, ssc[h][k]);
#pragma unroll
  for (int off = 16; off; off >>= 1) mx = fmaxf(mx, __shfl_xor(mx, off, 32));
  float sum = 0.f;
  for (int k = lane; k < SS; k += 32) {
    float e = __expf(ssc[h][k] - mx);
    ssc[h][k] = e;
    sum += e;
  }
#pragma unroll
  for (int off = 16; off; off >>= 1) sum += __shfl_xor(sum, off, 32);
  __syncthreads();

  const float rinv = 1.f / sum;
  const int d0 = lane * 2;
  float a0 = 0.f, a1 = 0.f;
  for (int k = 0; k < SS; ++k) {
    float w = ssc[h][k];
    float2 vv = *(const float2*)(V + ((size_t)(b * SS + k) * NHH + h) * DD + d0);
    float2 ev = *(const float2*)(EV + (bq * SS + k) * DD + d0);
    a0 += w * (vv.x + ev.x);
    a1 += w * (vv.y + ev.y);
  }
  Xout[bq * HH + h * DD + d0]     = a0 * rinv;
  Xout[bq * HH + h * DD + d0 + 1] = a1 * rinv;
}

// ---------------------------------------------------------------------------
// Sparse attention rows (q >= 3): keep-mask admits only keys {0,1,2,g,g+1}.
// One block per (b,q); wave = head. Lanes 0..4 score one candidate each.
// If all 5 candidates are graph-masked, reference softmax degenerates to
// uniform 1/S over ALL keys -> use precomputed vsum + full edge_value row.
// ---------------------------------------------------------------------------
__global__ __launch_bounds__(256) void attn_sparse_kernel(
    const float* __restrict__ Q, const float* __restrict__ K,
    const float* __restrict__ V, const unsigned char* __restrict__ G,
    const float* __restrict__ EK, const float* __restrict__ EV,
    const float* __restrict__ EQ, const float* __restrict__ VS,
    float* __restrict__ Xout)
{
  const int b  = blockIdx.x / (SS - 3);
  const int qi = 3 + blockIdx.x % (SS - 3);
  const size_t bq = (size_t)b * SS + qi;
  const int h = threadIdx.x >> 5, lane = threadIdx.x & 31;
  const int g0 = 3 + (((qi - 3) >> 1) << 1);
  const int cand[5] = {0, 1, 2, g0, g0 + 1};

  float s = -3.4e38f;
  bool valid = false;
  if (lane < 5) {
    int k = cand[lane];
    const float* qh   = Q + bq * HH + h * DD;
    const float* krow = K + ((size_t)(b * SS + k) * NHH + h) * DD;
    const float* ekr  = EK + (bq * SS + k) * DD;
    const float* eqr  = EQ + (((size_t)b * SS + k) * SS + qi) * DD;
    float acc = 0.f;
#pragma unroll 8
    for (int d = 0; d < DD; ++d)
      acc += (qh[d] + eqr[d]) * (krow[d] + ekr[d]);
    valid = (G[((size_t)b * SS + qi) * SS + k] != 0);
    s = valid ? acc * 0.125f : -100000.f;
  }
  float mx = s;
#pragma unroll
  for (int off = 16; off; off >>= 1) mx = fmaxf(mx, __shfl_xor(mx, off, 32));
  float e = (lane < 5) ? __expf(s - mx) : 0.f;
  float sum = e;
#pragma unroll
  for (int off = 16; off; off >>= 1) sum += __shfl_xor(sum, off, 32);
  const bool anyvalid = (__ballot(valid) != 0ull);

  const int d0 = lane * 2;
  float a0 = 0.f, a1 = 0.f;
  if (anyvalid) {
    const float rinv = 1.f / sum;
#pragma unroll
    for (int j = 0; j < 5; ++j) {
      float w = __shfl(e, j, 32) * rinv;
      int k = cand[j];
      float2 vv = *(const float2*)(V + ((size_t)(b * SS + k) * NHH + h) * DD + d0);
      float2 ev = *(const float2*)(EV + (bq * SS + k) * DD + d0);
      a0 += w * (vv.x + ev.x);
      a1 += w * (vv.y + ev.y);
    }
  } else {                                       // uniform 1/S over all keys
    for (int k = 0; k < SS; ++k) {
      float2 ev = *(const float2*)(EV + (bq * SS + k) * DD + d0);
      a0 += ev.x; a1 += ev.y;
    }
    a0 += VS[b * HH + h * DD + d0];
    a1 += VS[b * HH + h * DD + d0 + 1];
    const float inv = 1.f / (float)SS;
    a0 *= inv; a1 *= inv;
  }
  Xout[bq * HH + h * DD + d0]     = a0;
  Xout[bq * HH + h * DD + d0 + 1] = a1;
}

// ---------------------------------------------------------------------------
extern "C" void kernel_launch(void* const* d_in, const int* in_sizes, int n_in,
                              void* d_out, int out_size, void* d_ws, size_t ws_size,
                              hipStream_t stream) {
  (void)in_sizes; (void)n_in; (void)out_size; (void)ws_size;
  const float*         query = (const float*)d_in[0];
  const float*         keyx  = (const float*)d_in[1];
  const float*         value = (const float*)d_in[2];
  const unsigned char* graph = (const unsigned char*)d_in[3];
  const float*         eK    = (const float*)d_in[4];
  const float*         eV    = (const float*)d_in[5];
  const float*         eQ    = (const float*)d_in[6];
  const float*         Wq    = (const float*)d_in[7];
  const float*         bq    = (const float*)d_in[8];
  const float*         Wk    = (const float*)d_in[9];
  const float*         bk    = (const float*)d_in[10];
  const float*         Wv    = (const float*)d_in[11];
  const float*         bv    = (const float*)d_in[12];
  const float*         Wo    = (const float*)d_in[13];
  const float*         bo    = (const float*)d_in[14];

  float* ws = (float*)d_ws;
  const size_t nbuf = (size_t)BSR * HH;          // 525312 floats per buffer
  float* qb   = ws;
  float* kb   = qb + nbuf;
  float* vb   = kb + nbuf;
  float* xb   = vb + nbuf;
  float* vsum = xb + nbuf;                       // BB*HH floats

  dim3 gpc(72, HH / 64);                         // classed: 18 subseq x 4 tiles
  dim3 gpp((BSR + 15) / 16, HH / 64);            // plain: 65 contiguous tiles
  dim3 bp(128);
  proj_wmma_kernel<<<gpc, bp, 0, stream>>>(query, Wq, bq, qb, 5);
  proj_wmma_kernel<<<gpc, bp, 0, stream>>>(keyx,  Wk, bk, kb, 5);
  proj_wmma_kernel<<<gpc, bp, 0, stream>>>(value, Wv, bv, vb, 5);
  vsum_kernel<<<dim3((BB * HH + 255) / 256), dim3(256), 0, stream>>>(vb, vsum);
  attn_dense_kernel<<<dim3(BB * 3), dim3(256), 0, stream>>>(
      qb, kb, vb, graph, eK, eV, eQ, xb);
  attn_sparse_kernel<<<dim3(BB * (SS - 3)), dim3(256), 0, stream>>>(
      qb, kb, vb, graph, eK, eV, eQ, vsum, xb);
  proj_wmma_kernel<<<gpp, bp, 0, stream>>>(xb, Wo, bo, (float*)d_out, 1);
}